// FlexiCubesGeometry_44229573214720
// MI455X (gfx1250) — compile-verified
//
#include <hip/hip_runtime.h>
#include <stdint.h>

// FlexiCubes core for MI455X (gfx1250, wave32).
// Memory-bound kernel: one thread per cell, weights tile staged to LDS via
// CDNA5 async global->LDS copies, deterministic two-stage reduction for reg.

#define RES     96
#define NCELLS  (RES * RES * RES)      // 884736
#define BLK     256                    // 8 wave32 waves
#define NBLK    (NCELLS / BLK)         // 3456 (exact)
#define NW      21                     // weights per cell
#define WDW     (BLK * NW)             // 5376 dwords staged per block

#if defined(__AMDGCN__) && \
    __has_builtin(__builtin_amdgcn_global_load_async_to_lds_b128) && \
    __has_builtin(__builtin_amdgcn_global_load_async_to_lds_b32)
#define USE_ASYNC_BUILTIN 1
#endif

// Exact parameter types for the async builtins (leaked by clang diagnostics):
//   b128: (v4i addrspace(1)*, v4i addrspace(3)*, imm int, imm int)
//   b32 : (int addrspace(1)*, int addrspace(3)*, imm int, imm int)
typedef int v4i __attribute__((ext_vector_type(4)));
typedef __attribute__((address_space(1))) v4i*  g_v4i_p;
typedef __attribute__((address_space(3))) v4i*  l_v4i_p;
typedef __attribute__((address_space(1))) int*  g_i_p;
typedef __attribute__((address_space(3))) int*  l_i_p;

__launch_bounds__(BLK)
__global__ void flexi_main(const float* __restrict__ verts,
                           const int*   __restrict__ indices,
                           const float* __restrict__ sdf,
                           const float* __restrict__ deform,
                           const float* __restrict__ weights,
                           float* __restrict__ vd_out,
                           float* __restrict__ blk_num,
                           unsigned int* __restrict__ blk_cnt)
{
    __shared__ __align__(16) float wsh[WDW];   // 21504 B weights tile
    __shared__ float        rsum[BLK];
    __shared__ unsigned int rcnt[BLK];

    const int tid  = threadIdx.x;
    const int blk  = blockIdx.x;
    const int cell = blk * BLK + tid;

    // ---------------- stage weights tile: global -> LDS (async) -------------
    // Block tile = BLK*21 floats = 21504 B, 16B-aligned.
    // 5 full rounds of b128 (dwords 0..5119) + one b32 tail (dwords 5120..5375).
    const float* gw = weights + (size_t)blk * WDW;
#if defined(USE_ASYNC_BUILTIN)
    {
        auto gsrc = (__attribute__((address_space(1))) char*)gw;
        auto ldst = (__attribute__((address_space(3))) char*)wsh;
#pragma unroll
        for (int k = 0; k < 5; ++k) {
            int i = k * BLK + tid;     // b128 chunk index
            __builtin_amdgcn_global_load_async_to_lds_b128(
                (g_v4i_p)(gsrc + (size_t)i * 16),
                (l_v4i_p)(ldst + i * 16),
                0, 0);
        }
        int t = 5120 + tid;            // tail dword index
        __builtin_amdgcn_global_load_async_to_lds_b32(
            (g_i_p)(gsrc + (size_t)t * 4),
            (l_i_p)(ldst + t * 4),
            0, 0);
    }
#else
    {
        unsigned lbase = (unsigned)(size_t)(__attribute__((address_space(3))) char*)wsh;
        uint64_t gbase = (uint64_t)(uintptr_t)gw;
#pragma unroll
        for (int k = 0; k < 5; ++k) {
            int i = k * BLK + tid;
            unsigned l = lbase + (unsigned)i * 16u;
            uint64_t g = gbase + (uint64_t)i * 16u;
            asm volatile("global_load_async_to_lds_b128 %0, %1, off"
                         :: "v"(l), "v"(g) : "memory");
        }
        unsigned l = lbase + (unsigned)(5120 + tid) * 4u;
        uint64_t g = gbase + (uint64_t)(5120 + tid) * 4u;
        asm volatile("global_load_async_to_lds_b32 %0, %1, off"
                     :: "v"(l), "v"(g) : "memory");
    }
#endif
#if defined(__AMDGCN__) && __has_builtin(__builtin_amdgcn_s_wait_asynccnt)
    __builtin_amdgcn_s_wait_asynccnt(0);
#else
    asm volatile("s_wait_asynccnt 0" ::: "memory");
#endif
    __syncthreads();

    // ---------------- gather corners ----------------------------------------
    const float MD = (float)(2.0 / 96.0 / 4.0);   // MAX_DISP as f32
    int ci[8];
    {
        const int4* ip4 = reinterpret_cast<const int4*>(indices + (size_t)cell * 8);
        int4 a = ip4[0], b = ip4[1];
        ci[0] = a.x; ci[1] = a.y; ci[2] = a.z; ci[3] = a.w;
        ci[4] = b.x; ci[5] = b.y; ci[6] = b.z; ci[7] = b.w;
    }

    float s[8], xx[8], xy[8], xz[8];
#pragma unroll
    for (int k = 0; k < 8; ++k) {
        int id = ci[k];
        s[k] = sdf[id];
        const float* vp = verts  + (size_t)id * 3;
        const float* dp = deform + (size_t)id * 3;
        xx[k] = vp[0] + MD * tanhf(dp[0]);
        xy[k] = vp[1] + MD * tanhf(dp[1]);
        xz[k] = vp[2] + MD * tanhf(dp[2]);
    }

    // weights record for this cell (LDS; stride 21 coprime with 64 banks)
    const float* wp = &wsh[tid * NW];
    float alpha[8];
#pragma unroll
    for (int k = 0; k < 8; ++k) alpha[k] = wp[12 + k];

    // ---------------- 12 edges ----------------------------------------------
    constexpr int EA[12] = {0, 1, 4, 0, 2, 3, 6, 2, 2, 3, 7, 6};
    constexpr int EB[12] = {1, 5, 5, 4, 3, 7, 7, 6, 0, 1, 5, 4};

    float px[12], py[12], pz[12];
    float wsum = 0.f, vx = 0.f, vy = 0.f, vz = 0.f;
    int crossmask = 0, ncross = 0;
#pragma unroll
    for (int e = 0; e < 12; ++e) {
        const int a = EA[e], b = EB[e];
        const float sa = s[a], sb = s[b];
        const bool cross = (sa > 0.f) != (sb > 0.f);
        float pex = 0.f, pey = 0.f, pez = 0.f, we = 0.f;
        if (cross) {
            const float t1 = alpha[a] * sb;
            const float t2 = alpha[b] * sa;
            const float den = t1 - t2;
            pex = (t1 * xx[a] - t2 * xx[b]) / den;
            pey = (t1 * xy[a] - t2 * xy[b]) / den;
            pez = (t1 * xz[a] - t2 * xz[b]) / den;
            we  = wp[e];                  // beta[e]
            crossmask |= (1 << e);
            ++ncross;
        }
        px[e] = pex; py[e] = pey; pz[e] = pez;
        wsum += we;
        vx += we * pex; vy += we * pey; vz += we * pez;
    }

    const bool surf = (ncross > 0);
    float vdx = 0.f, vdy = 0.f, vdz = 0.f;
    if (surf) { vdx = vx / wsum; vdy = vy / wsum; vdz = vz / wsum; }

    float devsum = 0.f;
#pragma unroll
    for (int e = 0; e < 12; ++e) {
        if (crossmask & (1 << e)) {
            const float dx = px[e] - vdx;
            const float dy = py[e] - vdy;
            const float dz = pz[e] - vdz;
            devsum += sqrtf(dx * dx + dy * dy + dz * dz + 1e-12f);
        }
    }
    const int   nc  = (ncross > 1) ? ncross : 1;
    const float dev = devsum / (float)nc;
    const float contrib = surf ? dev * wp[20] : 0.f;

    // ---------------- outputs -----------------------------------------------
    float* o = vd_out + (size_t)cell * 3;
    o[0] = vdx; o[1] = vdy; o[2] = vdz;

    // deterministic per-block reduction (fixed-order LDS tree)
    rsum[tid] = contrib;
    rcnt[tid] = surf ? 1u : 0u;
    __syncthreads();
#pragma unroll
    for (int off = BLK / 2; off > 0; off >>= 1) {
        if (tid < off) {
            rsum[tid] += rsum[tid + off];
            rcnt[tid] += rcnt[tid + off];
        }
        __syncthreads();
    }
    if (tid == 0) {
        blk_num[blk] = rsum[0];
        blk_cnt[blk] = rcnt[0];
    }
}

// Deterministic final reduction over NBLK partials, one block.
__launch_bounds__(BLK)
__global__ void flexi_reduce(const float* __restrict__ blk_num,
                             const unsigned int* __restrict__ blk_cnt,
                             float* __restrict__ reg_out)
{
    __shared__ float        sn[BLK];
    __shared__ unsigned int sc[BLK];
    float a = 0.f; unsigned c = 0;
    for (int i = threadIdx.x; i < NBLK; i += BLK) {   // fixed order
        a += blk_num[i];
        c += blk_cnt[i];
    }
    sn[threadIdx.x] = a;
    sc[threadIdx.x] = c;
    __syncthreads();
#pragma unroll
    for (int off = BLK / 2; off > 0; off >>= 1) {
        if (threadIdx.x < off) {
            sn[threadIdx.x] += sn[threadIdx.x + off];
            sc[threadIdx.x] += sc[threadIdx.x + off];
        }
        __syncthreads();
    }
    if (threadIdx.x == 0) {
        const unsigned n = (sc[0] > 1u) ? sc[0] : 1u;
        reg_out[0] = sn[0] / (float)n;
    }
}

extern "C" void kernel_launch(void* const* d_in, const int* in_sizes, int n_in,
                              void* d_out, int out_size, void* d_ws, size_t ws_size,
                              hipStream_t stream)
{
    (void)in_sizes; (void)n_in; (void)out_size; (void)ws_size;
    const float* verts   = (const float*)d_in[0];
    const int*   indices = (const int*)  d_in[1];
    const float* sdf     = (const float*)d_in[2];
    const float* deform  = (const float*)d_in[3];
    const float* weights = (const float*)d_in[4];

    float* vd_out  = (float*)d_out;                    // (RES^3, 3)
    float* reg_out = vd_out + (size_t)NCELLS * 3;      // scalar tail

    float*        blk_num = (float*)d_ws;
    unsigned int* blk_cnt = (unsigned int*)((char*)d_ws + (size_t)NBLK * sizeof(float));

    flexi_main<<<NBLK, BLK, 0, stream>>>(verts, indices, sdf, deform, weights,
                                         vd_out, blk_num, blk_cnt);
    flexi_reduce<<<1, BLK, 0, stream>>>(blk_num, blk_cnt, reg_out);
}